// SANLayer_69148973465916
// MI455X (gfx1250) — compile-verified
//
#include <hip/hip_runtime.h>
#include <math.h>

typedef __attribute__((ext_vector_type(2))) float v2f;
typedef __attribute__((ext_vector_type(8))) float v8f;

#define NN   4096
#define CIN  128
#define COUT 128

// workspace layout (float offsets)
#define OFF_V    0            // 20*128 pre-contracted weight vectors
#define OFF_D    2560         // 2*2*5*4096 per-node dots (s=0..3 att segs, s=4 ones)
#define OFF_SRC  84480        // 2*4096
#define OFF_DST  92672        // 2*4096
#define OFF_M    100864       // 2*4096 row max
#define OFF_SINV 109056       // 2*4096 1/rowsum
#define OFF_C    117248       // 2*4096 colsum(alpha)
#define OFF_C2   125440       // 2*4096 colsum(alpha^2)
#define OFF_Z    133632       // 1 scalar (z_irr + z_sol)
#define OFF_Y    133664       // 4096*128  y = x @ W_har

static __device__ __forceinline__ float lrelu(float z) {
    return z >= 0.f ? z : 0.01f * z;
}

// v[(b*2+j)*5+s][i] = sum_o W_b[j][i][o] * att_seg_s[o]   (s=4 -> ones)
__global__ void k_precontract(const float* __restrict__ wIrr, const float* __restrict__ wSol,
                              const float* __restrict__ aIrr, const float* __restrict__ aSol,
                              float* __restrict__ ws) {
    int idx = blockIdx.x;               // 0..19 = b*10 + j*5 + s
    int b = idx / 10;
    int j = (idx / 5) % 2;
    int s = idx % 5;
    const float* W = (b == 0 ? wIrr : wSol) + j * CIN * COUT;
    const float* A = (b == 0 ? aIrr : aSol);
    int i = threadIdx.x;                // 0..127 (CIN index)
    float acc = 0.f;
    for (int o = 0; o < COUT; ++o) {
        float a = (s < 4) ? A[s * COUT + o] : 1.0f;
        acc += W[i * COUT + o] * a;
    }
    ws[OFF_V + idx * CIN + i] = acc;
}

// d[v][n] = x[n,:] . vvec[v]   for all 20 vectors
__global__ void k_node_dots(const float* __restrict__ x, float* __restrict__ ws) {
    __shared__ float red[8];
    int n = blockIdx.x;
    int tid = threadIdx.x;              // 128 threads
    int lane = tid & 31, wid = tid >> 5;
    float xi = x[(size_t)n * CIN + tid];
    for (int v = 0; v < 20; ++v) {
        float p = xi * ws[OFF_V + v * CIN + tid];
        for (int o = 16; o > 0; o >>= 1) p += __shfl_xor(p, o, 32);
        if (lane == 0) red[wid] = p;
        __syncthreads();
        if (tid == 0) ws[OFF_D + (size_t)v * NN + n] = red[0] + red[1] + red[2] + red[3];
        __syncthreads();
    }
}

// replicate the torch reshape: hf row r mixes nodes 2r' and 2r'+1 of branch j=(r>=2048)
__global__ void k_build_srcdst(float* __restrict__ ws) {
    int idx = blockIdx.x * blockDim.x + threadIdx.x;
    if (idx >= 2 * NN) return;
    int b = idx >> 12;
    int r = idx & (NN - 1);
    int j = (r >= NN / 2) ? 1 : 0;
    int rp = r & (NN / 2 - 1);
    const float* d = ws + OFF_D + (size_t)(b * 2 + j) * 5 * NN;
    ws[OFF_SRC + b * NN + r] = d[0 * NN + 2 * rp] + d[1 * NN + 2 * rp + 1];
    ws[OFF_DST + b * NN + r] = d[2 * NN + 2 * rp] + d[3 * NN + 2 * rp + 1];
}

// per-row masked softmax stats: M[n] = max, Sinv[n] = 1/sum(exp(z-M))
__global__ void k_rowstats(const float* __restrict__ mask, const float* __restrict__ src,
                           const float* __restrict__ dst,
                           float* __restrict__ Mrow, float* __restrict__ Sinv) {
    __shared__ float zbuf[NN];          // 16 KB of the 320 KB/WGP LDS
    __shared__ float red[8];
    int n = blockIdx.x;
    int tid = threadIdx.x;              // 128
    int lane = tid & 31, wid = tid >> 5;
    float sn = src[n];
    float mx = -INFINITY;
    for (int m = tid; m < NN; m += 128) {
        float on = mask[(size_t)n * NN + m];
        float z = -INFINITY;
        if (on > 0.f) { z = lrelu(sn + dst[m]); mx = fmaxf(mx, z); }
        zbuf[m] = z;
    }
    for (int o = 16; o > 0; o >>= 1) mx = fmaxf(mx, __shfl_xor(mx, o, 32));
    if (lane == 0) red[wid] = mx;
    __syncthreads();
    if (tid == 0) red[4] = fmaxf(fmaxf(red[0], red[1]), fmaxf(red[2], red[3]));
    __syncthreads();
    float M = red[4];
    float Mw = (M > -INFINITY) ? M : 0.f;
    float s = 0.f;
    for (int m = tid; m < NN; m += 128) {
        float z = zbuf[m];
        if (z > -INFINITY) s += __expf(z - Mw);
    }
    for (int o = 16; o > 0; o >>= 1) s += __shfl_xor(s, o, 32);
    if (lane == 0) red[wid] = s;
    __syncthreads();
    if (tid == 0) {
        float S = red[0] + red[1] + red[2] + red[3];
        if (S == 0.f) S = 1.f;
        Mrow[n] = Mw;
        Sinv[n] = 1.f / S;
    }
}

// c[m] += sum_n alpha[n][m]   (column sums, coalesced column stripes)
__global__ void k_colsum(const float* __restrict__ mask, const float* __restrict__ src,
                         const float* __restrict__ dst, const float* __restrict__ Mrow,
                         const float* __restrict__ Sinv, float* __restrict__ c) {
    int m = blockIdx.x * 256 + threadIdx.x;
    float dm = dst[m];
    float acc = 0.f;
    int n0 = blockIdx.y * 256;
    for (int n = n0; n < n0 + 256; ++n) {
        float on = mask[(size_t)n * NN + m];
        if (on > 0.f) {
            float z = lrelu(src[n] + dm);
            acc += __expf(z - Mrow[n]) * Sinv[n];
        }
    }
    atomicAdd(&c[m], acc);
}

// c2[m] += sum_n cw[n] * alpha[n][m]  == colsum(alpha^2) given cw = colsum(alpha)
__global__ void k_colsum2(const float* __restrict__ mask, const float* __restrict__ src,
                          const float* __restrict__ dst, const float* __restrict__ Mrow,
                          const float* __restrict__ Sinv, const float* __restrict__ cw,
                          float* __restrict__ c2) {
    int m = blockIdx.x * 256 + threadIdx.x;
    float dm = dst[m];
    float acc = 0.f;
    int n0 = blockIdx.y * 256;
    for (int n = n0; n < n0 + 256; ++n) {
        float on = mask[(size_t)n * NN + m];
        if (on > 0.f) {
            float z = lrelu(src[n] + dm);
            acc += cw[n] * __expf(z - Mrow[n]) * Sinv[n];
        }
    }
    atomicAdd(&c2[m], acc);
}

// z += c . rs0 + c2 . rs1
__global__ void k_branchdot(const float* __restrict__ c, const float* __restrict__ c2,
                            const float* __restrict__ rs0, const float* __restrict__ rs1,
                            float* __restrict__ z) {
    __shared__ float red[8];
    int tid = threadIdx.x;              // 256
    int lane = tid & 31, wid = tid >> 5;
    float acc = 0.f;
    for (int m = tid; m < NN; m += 256)
        acc += c[m] * rs0[m] + c2[m] * rs1[m];
    for (int o = 16; o > 0; o >>= 1) acc += __shfl_xor(acc, o, 32);
    if (lane == 0) red[wid] = acc;
    __syncthreads();
    if (tid == 0) {
        float t = 0.f;
        for (int w = 0; w < 8; ++w) t += red[w];
        atomicAdd(z, t);
    }
}

__global__ void k_zero(float* __restrict__ p, int count) {
    int i = blockIdx.x * blockDim.x + threadIdx.x;
    if (i < count) p[i] = 0.f;
}

// C[M x N] = A[M x K] @ B[K x N] (+ *addScalar), f32 WMMA 16x16x4, one wave per 16x16 tile.
// 8 waves per block; tile count must be a multiple of 8 (it is: 2048 for both uses).
__global__ void k_gemm_wmma(const float* __restrict__ A, int lda,
                            const float* __restrict__ B, int ldb,
                            float* __restrict__ C, int ldc,
                            int K, int nTilesN, const float* __restrict__ addScalar) {
    int lane = threadIdx.x & 31;
    int wid  = threadIdx.x >> 5;
    int t = blockIdx.x * 8 + wid;
    int tileN = t % nTilesN;
    int tileM = t / nTilesN;
    int half = lane >> 4;               // K-half select (A/B layout, ISA 7.12.2)
    int lr   = lane & 15;
    const float* arow = A + (size_t)(tileM * 16 + lr) * lda + 2 * half;
    const float* bcol = B + (size_t)(2 * half) * ldb + tileN * 16 + lr;
    v8f acc = {0.f, 0.f, 0.f, 0.f, 0.f, 0.f, 0.f, 0.f};
    for (int k = 0; k < K; k += 4) {
        __builtin_prefetch(arow + k + 64, 0, 0);    // global_prefetch_b8
        float2 av = *(const float2*)(arow + k);     // contiguous K-pair in row
        v2f a; a.x = av.x; a.y = av.y;
        v2f b; b.x = bcol[(size_t)k * ldb];
               b.y = bcol[(size_t)(k + 1) * ldb];
        // 8-arg form: (neg_a, A, neg_b, B, c_mod, C, reuse_a, reuse_b)
        acc = __builtin_amdgcn_wmma_f32_16x16x4_f32(false, a, false, b,
                                                    (short)0, acc, false, false);
    }
    float zadd = addScalar ? addScalar[0] : 0.f;
    float* crow = C + (size_t)(tileM * 16 + 8 * half) * ldc + tileN * 16 + lr;
    for (int v = 0; v < 8; ++v)
        crow[(size_t)v * ldc] = acc[v] + zadd;      // D VGPR v -> row v (+8*half)
}

extern "C" void kernel_launch(void* const* d_in, const int* in_sizes, int n_in,
                              void* d_out, int out_size, void* d_ws, size_t ws_size,
                              hipStream_t stream) {
    const float* x    = (const float*)d_in[0];
    const float* Lup  = (const float*)d_in[1];
    const float* Ldn  = (const float*)d_in[2];
    const float* P    = (const float*)d_in[3];
    const float* wIrr = (const float*)d_in[4];
    const float* wSol = (const float*)d_in[5];
    const float* wHar = (const float*)d_in[6];
    const float* aIrr = (const float*)d_in[7];
    const float* aSol = (const float*)d_in[8];
    float* out = (float*)d_out;
    float* ws  = (float*)d_ws;

    // zero accumulators: c, c2, z (contiguous region)
    k_zero<<<(2 * 8192 + 1 + 255) / 256, 256, 0, stream>>>(ws + OFF_C, 2 * 8192 + 1);

    k_precontract<<<20, 128, 0, stream>>>(wIrr, wSol, aIrr, aSol, ws);
    k_node_dots<<<NN, 128, 0, stream>>>(x, ws);
    k_build_srcdst<<<(2 * NN + 255) / 256, 256, 0, stream>>>(ws);

    // y = x @ W_har  (2048 tiles -> 256 blocks of 8 waves)
    k_gemm_wmma<<<256, 256, 0, stream>>>(x, CIN, wHar, COUT, ws + OFF_Y, COUT,
                                         CIN, COUT / 16, nullptr);

    const float* masks[2] = { Ldn, Lup };   // branch 0 = irr/Ldown, branch 1 = sol/Lup
    for (int b = 0; b < 2; ++b) {
        const float* src = ws + OFF_SRC + b * NN;
        const float* dst = ws + OFF_DST + b * NN;
        float* Mrow = ws + OFF_M + b * NN;
        float* Sinv = ws + OFF_SINV + b * NN;
        float* c    = ws + OFF_C + b * NN;
        float* c2   = ws + OFF_C2 + b * NN;
        k_rowstats<<<NN, 128, 0, stream>>>(masks[b], src, dst, Mrow, Sinv);
        k_colsum<<<dim3(16, 16), 256, 0, stream>>>(masks[b], src, dst, Mrow, Sinv, c);
        k_colsum2<<<dim3(16, 16), 256, 0, stream>>>(masks[b], src, dst, Mrow, Sinv, c, c2);
        const float* rs0 = ws + OFF_D + (size_t)((b * 2 + 0) * 5 + 4) * NN;
        const float* rs1 = ws + OFF_D + (size_t)((b * 2 + 1) * 5 + 4) * NN;
        k_branchdot<<<1, 256, 0, stream>>>(c, c2, rs0, rs1, ws + OFF_Z);
    }

    // out = P @ y + (z_irr + z_sol)   (2048 tiles -> 256 blocks of 8 waves)
    k_gemm_wmma<<<256, 256, 0, stream>>>(P, NN, ws + OFF_Y, COUT, out, COUT,
                                         NN, COUT / 16, ws + OFF_Z);
}